// SMPL_82815559401981
// MI455X (gfx1250) — compile-verified
//
#include <hip/hip_runtime.h>
#include <math.h>

#define NVERTS 6890
#define NJ 24
#define NBETAS 10
#define NBETAS_PAD 12
#define NPOSE 207
#define NPOSE_PAD 208
#define NCOCO 19
#define BATCH 1024
#define VCOLS (NVERTS*3)   // 20670

typedef __attribute__((ext_vector_type(2))) float v2f;
typedef __attribute__((ext_vector_type(8))) float v8f;

__device__ __constant__ int kParents[NJ] =
    {0,0,0,0,1,2,3,4,5,6,7,8,9,9,9,12,13,14,16,17,18,19,20,21};

__device__ __forceinline__ int imin(int a, int b) { return a < b ? a : b; }

// ---------------------------------------------------------------------------
// K0: Jdirs[j][d][k] = sum_v Jreg[j][v]*shapedirs[k][v*3+d];  Jtemp = Jreg @ v_template
// grid = 72 blocks (j*3+d), 256 threads
// ---------------------------------------------------------------------------
__global__ void jdirs_kernel(const float* __restrict__ Jreg,
                             const float* __restrict__ shapedirs,
                             const float* __restrict__ v_template,
                             float* __restrict__ jdirs, float* __restrict__ jtemp) {
    const int j = blockIdx.x / 3, d = blockIdx.x % 3;
    __shared__ float red[11][256];
    float acc[11];
#pragma unroll
    for (int k = 0; k < 11; ++k) acc[k] = 0.f;
    for (int v = threadIdx.x; v < NVERTS; v += 256) {
        const float w = Jreg[j * NVERTS + v];
#pragma unroll
        for (int k = 0; k < NBETAS; ++k)
            acc[k] += w * shapedirs[k * VCOLS + v * 3 + d];
        acc[10] += w * v_template[v * 3 + d];
    }
#pragma unroll
    for (int k = 0; k < 11; ++k) red[k][threadIdx.x] = acc[k];
    __syncthreads();
    for (int s = 128; s > 0; s >>= 1) {
        if ((int)threadIdx.x < s)
#pragma unroll
            for (int k = 0; k < 11; ++k) red[k][threadIdx.x] += red[k][threadIdx.x + s];
        __syncthreads();
    }
    if (threadIdx.x == 0) {
#pragma unroll
        for (int k = 0; k < NBETAS; ++k) jdirs[(j * 3 + d) * NBETAS + k] = red[k][0];
        jtemp[j * 3 + d] = red[10][0];
    }
}

// ---------------------------------------------------------------------------
// K0b: zero-padded beta for clean K=12 WMMA stepping (cols 10,11 = 0)
// ---------------------------------------------------------------------------
__global__ void pad_beta_kernel(const float* __restrict__ beta,
                                float* __restrict__ beta_pad) {
    const int idx = blockIdx.x * blockDim.x + threadIdx.x;
    if (idx >= BATCH * NBETAS_PAD) return;
    const int b = idx / NBETAS_PAD, k = idx % NBETAS_PAD;
    beta_pad[idx] = (k < NBETAS) ? beta[b * NBETAS + k] : 0.f;
}

// ---------------------------------------------------------------------------
// K1: Rodrigues. thread per (b, joint). Writes Rs (output) + padded pose_feature.
// ---------------------------------------------------------------------------
__global__ void rodrigues_kernel(const float* __restrict__ theta,
                                 float* __restrict__ Rs_out,
                                 float* __restrict__ pf) {
    const int idx = blockIdx.x * blockDim.x + threadIdx.x;
    if (idx >= BATCH * NJ) return;
    const int b = idx / NJ, i = idx % NJ;
    const float t0 = theta[b * 72 + i * 3 + 0];
    const float t1 = theta[b * 72 + i * 3 + 1];
    const float t2 = theta[b * 72 + i * 3 + 2];
    const float e0 = t0 + 1e-8f, e1 = t1 + 1e-8f, e2 = t2 + 1e-8f;
    const float ang = sqrtf(e0 * e0 + e1 * e1 + e2 * e2);
    const float inv = 1.f / ang;
    const float rx = t0 * inv, ry = t1 * inv, rz = t2 * inv;
    const float c = cosf(ang), s = sinf(ang), ic = 1.f - c;
    float K[3][3] = {{0.f, -rz, ry}, {rz, 0.f, -rx}, {-ry, rx, 0.f}};
    float R[3][3];
#pragma unroll
    for (int p = 0; p < 3; ++p)
#pragma unroll
        for (int q = 0; q < 3; ++q) {
            float kk = 0.f;
#pragma unroll
            for (int m = 0; m < 3; ++m) kk += K[p][m] * K[m][q];
            R[p][q] = (p == q ? 1.f : 0.f) + s * K[p][q] + ic * kk;
        }
    float* ro = Rs_out + b * (NJ * 9) + i * 9;
#pragma unroll
    for (int p = 0; p < 3; ++p)
#pragma unroll
        for (int q = 0; q < 3; ++q) ro[p * 3 + q] = R[p][q];
    if (i > 0) {
        float* po = pf + (size_t)b * NPOSE_PAD + (i - 1) * 9;
#pragma unroll
        for (int p = 0; p < 3; ++p)
#pragma unroll
            for (int q = 0; q < 3; ++q) po[p * 3 + q] = R[p][q] - (p == q ? 1.f : 0.f);
    } else {
        pf[(size_t)b * NPOSE_PAD + NPOSE] = 0.f;  // K-pad column
    }
}

// ---------------------------------------------------------------------------
// K1b: J[b][j][d] = Jtemp[j][d] + Jdirs[j][d][:] . beta[b]
// ---------------------------------------------------------------------------
__global__ void jreg_kernel(const float* __restrict__ beta,
                            const float* __restrict__ jdirs,
                            const float* __restrict__ jtemp,
                            float* __restrict__ J) {
    const int idx = blockIdx.x * blockDim.x + threadIdx.x;
    if (idx >= BATCH * 72) return;
    const int b = idx / 72, jd = idx % 72;
    float acc = jtemp[jd];
#pragma unroll
    for (int k = 0; k < NBETAS; ++k) acc += jdirs[jd * NBETAS + k] * beta[b * NBETAS + k];
    J[b * 72 + jd] = acc;
}

// ---------------------------------------------------------------------------
// K2: kinematic chain. 1 thread per batch element, per-thread LDS slice for
// the 24 global (R|t) results. Emits A[b][24][16].
// ---------------------------------------------------------------------------
__global__ void kinematics_kernel(const float* __restrict__ Rs,
                                  const float* __restrict__ J,
                                  float* __restrict__ A) {
    const int b = blockIdx.x * 32 + threadIdx.x;
    __shared__ float res[32][NJ][12];  // [thread][joint][R(9)+t(3)]
    if (b >= BATCH) return;
    float(*my)[12] = res[threadIdx.x];
    const float* Jb = J + b * 72;
    for (int i = 0; i < NJ; ++i) {
        float R[9];
#pragma unroll
        for (int e = 0; e < 9; ++e) R[e] = Rs[b * (NJ * 9) + i * 9 + e];
        float Rg[9], tg[3];
        if (i == 0) {
            // root_rot = R @ diag(1,-1,-1): negate columns 1,2
#pragma unroll
            for (int p = 0; p < 3; ++p) {
                Rg[p * 3 + 0] = R[p * 3 + 0];
                Rg[p * 3 + 1] = -R[p * 3 + 1];
                Rg[p * 3 + 2] = -R[p * 3 + 2];
            }
#pragma unroll
            for (int p = 0; p < 3; ++p) tg[p] = Jb[p];
        } else {
            const int par = kParents[i];
            float tr[3];
#pragma unroll
            for (int p = 0; p < 3; ++p) tr[p] = Jb[i * 3 + p] - Jb[par * 3 + p];
            const float* Rp = my[par];
#pragma unroll
            for (int p = 0; p < 3; ++p) {
#pragma unroll
                for (int q = 0; q < 3; ++q) {
                    float acc = 0.f;
#pragma unroll
                    for (int m = 0; m < 3; ++m) acc += Rp[p * 3 + m] * R[m * 3 + q];
                    Rg[p * 3 + q] = acc;
                }
                float acc = Rp[9 + p];
#pragma unroll
                for (int m = 0; m < 3; ++m) acc += Rp[p * 3 + m] * tr[m];
                tg[p] = acc;
            }
        }
#pragma unroll
        for (int e = 0; e < 9; ++e) my[i][e] = Rg[e];
#pragma unroll
        for (int p = 0; p < 3; ++p) my[i][9 + p] = tg[p];
        // A = [[Rg, tg - Rg@J_i],[0 0 0 1]]
        float* Ab = A + ((size_t)b * NJ + i) * 16;
#pragma unroll
        for (int p = 0; p < 3; ++p) {
            float ta = 0.f;
#pragma unroll
            for (int q = 0; q < 3; ++q) {
                Ab[p * 4 + q] = Rg[p * 3 + q];
                ta += Rg[p * 3 + q] * Jb[i * 3 + q];
            }
            Ab[p * 4 + 3] = tg[p] - ta;
        }
        Ab[12] = 0.f; Ab[13] = 0.f; Ab[14] = 0.f; Ab[15] = 1.f;
    }
}

// ---------------------------------------------------------------------------
// K3: fused shape+pose GEMM + WMMA LBS blend. One wave per (16 batch x 16 vert)
// tile. C layout: reg r, lane l -> (b = btile*16 + r + 8*(l>=16), v = vtile*16 + l%16).
// A-frag: lane l holds (m = l%16, k = r + 2*(l/16)); B-frag: (k = r + 2*(l/16), n = l%16).
// K-padding lives entirely on the A side (beta_pad col 10-11 = 0, pf col 207 = 0),
// so B-row addresses are merely clamped -> all loads are unconditional/branchless.
// ---------------------------------------------------------------------------
__global__ __launch_bounds__(256)
void lbs_kernel(const float* __restrict__ beta_pad, const float* __restrict__ shapedirs,
                const float* __restrict__ posedirs, const float* __restrict__ v_template,
                const float* __restrict__ weights, const float* __restrict__ pf,
                const float* __restrict__ A, float* __restrict__ verts) {
    const int wave = threadIdx.x >> 5;
    const int lane = threadIdx.x & 31;
    const int half = lane >> 4;
    const int l16 = lane & 15;
    const int NVT = (NVERTS + 15) / 16;  // 431
    const int vtile = blockIdx.x * 8 + wave;
    if (vtile >= NVT) return;
    const int btile = blockIdx.y;
    const int b_a = btile * 16 + l16;         // batch row for A-fragments
    const int v = vtile * 16 + l16;           // vertex column (N lane)
    const int vc = v < NVERTS ? v : NVERTS - 1;
    const int vcol = vc * 3;                  // base column in shapedirs/posedirs

    v8f acc[3] = {{}, {}, {}};  // x,y,z of v_posed for this tile

    // ---- GEMM1: beta_pad(16x12) @ shapedirs, tile t <-> component d ----
#pragma unroll
    for (int kk = 0; kk < 3; ++kk) {
        const int k0 = kk * 4 + 2 * half;
        v2f a;
        a.x = beta_pad[b_a * NBETAS_PAD + k0];
        a.y = beta_pad[b_a * NBETAS_PAD + k0 + 1];
        const int r0 = imin(k0, NBETAS - 1);       // A col >=10 is zero; clamp row
        const int r1 = imin(k0 + 1, NBETAS - 1);
#pragma unroll
        for (int t = 0; t < 3; ++t) {
            v2f bb;
            bb.x = shapedirs[r0 * VCOLS + vcol + t];
            bb.y = shapedirs[r1 * VCOLS + vcol + t];
            acc[t] = __builtin_amdgcn_wmma_f32_16x16x4_f32(
                false, a, false, bb, (short)0, acc[t], false, false);
        }
    }
    // template add (broadcast over batch rows)
#pragma unroll
    for (int t = 0; t < 3; ++t) acc[t] += v_template[vcol + t];

    // ---- GEMM2: pose_feature(16x208) @ posedirs ----
#pragma unroll
    for (int kk = 0; kk < 52; ++kk) {
        const int k0 = kk * 4 + 2 * half;
        v2f a;
        a.x = pf[(size_t)b_a * NPOSE_PAD + k0];
        a.y = pf[(size_t)b_a * NPOSE_PAD + k0 + 1];  // pad col 207 is zero
        const size_t r0 = (size_t)imin(k0, NPOSE - 1) * VCOLS;  // A col 207 is zero
        const size_t r1 = (size_t)imin(k0 + 1, NPOSE - 1) * VCOLS;
#pragma unroll
        for (int t = 0; t < 3; ++t) {
            v2f bb;
            bb.x = posedirs[r0 + vcol + t];
            bb.y = posedirs[r1 + vcol + t];
            acc[t] = __builtin_amdgcn_wmma_f32_16x16x4_f32(
                false, a, false, bb, (short)0, acc[t], false, false);
        }
    }

    // ---- LBS blend via WMMA: T_pq[b,v] = sum_j A[b,j,p,q] * w^T[j,v], K=24 ----
    v2f wf[6];
#pragma unroll
    for (int kk = 0; kk < 6; ++kk) {
        const int j0 = kk * 4 + 2 * half;
        wf[kk].x = weights[vc * NJ + j0];
        wf[kk].y = weights[vc * NJ + j0 + 1];
    }
    v8f out[3];
#pragma unroll
    for (int p = 0; p < 3; ++p) {
        v8f T[4];
#pragma unroll
        for (int q = 0; q < 4; ++q) {
            v8f tacc = {};
#pragma unroll
            for (int kk = 0; kk < 6; ++kk) {
                const int j0 = kk * 4 + 2 * half;
                v2f a;
                a.x = A[((size_t)b_a * NJ + j0) * 16 + p * 4 + q];
                a.y = A[((size_t)b_a * NJ + j0 + 1) * 16 + p * 4 + q];
                tacc = __builtin_amdgcn_wmma_f32_16x16x4_f32(
                    false, a, false, wf[kk], (short)0, tacc, false, false);
            }
            T[q] = tacc;
        }
        out[p] = T[0] * acc[0] + T[1] * acc[1] + T[2] * acc[2] + T[3];
    }

    if (v < NVERTS) {
#pragma unroll
        for (int r = 0; r < 8; ++r) {
            const int b = btile * 16 + r + 8 * half;
            float* o = verts + ((size_t)b * NVERTS + v) * 3;
            o[0] = out[0][r];
            o[1] = out[1][r];
            o[2] = out[2][r];
        }
    }
}

// ---------------------------------------------------------------------------
// K4: joints = joint_regressor @ verts. Block per (b, coco joint).
// ---------------------------------------------------------------------------
__global__ void joints_kernel(const float* __restrict__ verts,
                              const float* __restrict__ jreg,
                              float* __restrict__ joints) {
    const int b = blockIdx.x / NCOCO;
    const int j = blockIdx.x % NCOCO;
    __shared__ float red[3][128];
    float ax = 0.f, ay = 0.f, az = 0.f;
    for (int v = threadIdx.x; v < NVERTS; v += 128) {
        const float w = jreg[j * NVERTS + v];
        const float* p = verts + ((size_t)b * NVERTS + v) * 3;
        ax += w * p[0];
        ay += w * p[1];
        az += w * p[2];
    }
    red[0][threadIdx.x] = ax; red[1][threadIdx.x] = ay; red[2][threadIdx.x] = az;
    __syncthreads();
    for (int s = 64; s > 0; s >>= 1) {
        if ((int)threadIdx.x < s) {
            red[0][threadIdx.x] += red[0][threadIdx.x + s];
            red[1][threadIdx.x] += red[1][threadIdx.x + s];
            red[2][threadIdx.x] += red[2][threadIdx.x + s];
        }
        __syncthreads();
    }
    if (threadIdx.x == 0) {
        float* o = joints + ((size_t)b * NCOCO + j) * 3;
        o[0] = red[0][0]; o[1] = red[1][0]; o[2] = red[2][0];
    }
}

// ---------------------------------------------------------------------------
extern "C" void kernel_launch(void* const* d_in, const int* in_sizes, int n_in,
                              void* d_out, int out_size, void* d_ws, size_t ws_size,
                              hipStream_t stream) {
    const float* beta       = (const float*)d_in[0];
    const float* theta      = (const float*)d_in[1];
    const float* v_template = (const float*)d_in[2];
    const float* shapedirs  = (const float*)d_in[3];
    const float* Jreg       = (const float*)d_in[4];
    const float* posedirs   = (const float*)d_in[5];
    const float* joint_reg  = (const float*)d_in[6];
    const float* weights    = (const float*)d_in[7];

    float* out    = (float*)d_out;
    float* verts  = out;                                   // 1024*6890*3
    float* joints = verts + (size_t)BATCH * NVERTS * 3;    // 1024*19*3
    float* Rs     = joints + (size_t)BATCH * NCOCO * 3;    // 1024*24*9

    float* ws       = (float*)d_ws;
    float* pf       = ws;                                  // 1024*208
    float* jdirs    = pf + (size_t)BATCH * NPOSE_PAD;      // 72*10
    float* jtemp    = jdirs + 72 * NBETAS;                 // 72
    float* J        = jtemp + 72;                          // 1024*72
    float* A        = J + (size_t)BATCH * 72;              // 1024*24*16
    float* beta_pad = A + (size_t)BATCH * NJ * 16;         // 1024*12

    jdirs_kernel<<<72, 256, 0, stream>>>(Jreg, shapedirs, v_template, jdirs, jtemp);
    pad_beta_kernel<<<(BATCH * NBETAS_PAD + 255) / 256, 256, 0, stream>>>(beta, beta_pad);
    rodrigues_kernel<<<(BATCH * NJ + 255) / 256, 256, 0, stream>>>(theta, Rs, pf);
    jreg_kernel<<<(BATCH * 72 + 255) / 256, 256, 0, stream>>>(beta, jdirs, jtemp, J);
    kinematics_kernel<<<BATCH / 32, 32, 0, stream>>>(Rs, J, A);

    dim3 grid((431 + 7) / 8, BATCH / 16);  // 54 x 64 blocks, 8 waves/block
    lbs_kernel<<<grid, 256, 0, stream>>>(beta_pad, shapedirs, posedirs, v_template,
                                         weights, pf, A, verts);

    joints_kernel<<<BATCH * NCOCO, 128, 0, stream>>>(verts, joint_reg, joints);
}